// TruncatedSignature_74088185856492
// MI455X (gfx1250) — compile-verified
//
#include <hip/hip_runtime.h>

// Depth-3 truncated path signature via three K=L GEMMs on V_WMMA_F32_16X16X4_F32.
//   S2[i,j]    = sum_t (pre[t,i]+dx[t,i]/2) * dx[t,j]
//   S3[i,j,k] += sum_t (pre[t,i]+dx[t,i]/3) * (dx[t,j]*dx[t,k]/2)      (8 x 64 GEMM)
//   S3[i,j,k] += sum_t (pre[t,i]+dx[t,i]/2)*dx[t,j] * suf[t,k]         (64 x 8 GEMM)
// with pre[t]=p[t]-p[0], dx[t]=p[t+1]-p[t], suf[t]=p[L-1]-p[t+1].

typedef __attribute__((ext_vector_type(2))) float v2f;
typedef __attribute__((ext_vector_type(4))) float v4f;
typedef __attribute__((ext_vector_type(8))) float v8f;

#define NCH 8
#define TC 256          // time steps per chunk per wave
#define NWAVES 4
#define SIGDIM 584      // 8 + 64 + 512
#define PART 576        // 64 + 512

// constant-index select-of-8 (used only outside hot loop)
#define SEL8(A,i) (((i)&4) ? (((i)&2) ? (((i)&1)?(A)[7]:(A)[6]) : (((i)&1)?(A)[5]:(A)[4])) \
                           : (((i)&2) ? (((i)&1)?(A)[3]:(A)[2]) : (((i)&1)?(A)[1]:(A)[0])))

__global__ __launch_bounds__(NWAVES * 32, 1)
void sig3_wmma_kernel(const float* __restrict__ path, float* __restrict__ out, int L)
{
    __shared__ __align__(16) float chunkBuf[NWAVES][TC + 2][NCH];
    __shared__ float partial[NWAVES][PART];

    const int b    = blockIdx.x;
    const int tid  = threadIdx.x;
    const int wave = tid >> 5;
    const int lane = tid & 31;
    const int m    = lane & 15;      // row/col index within 16x16 tile
    const int hi   = lane >> 4;      // 0 -> K slots 0,1 ; 1 -> K slots 2,3
    const int k7   = m & 7;
    const bool mlo = (m < 8);
    const bool mb  = (m & 8) != 0;

    const float* Pb = path + (size_t)b * NCH * L;

    float p0[NCH], pLv[NCH];
#pragma unroll
    for (int c = 0; c < NCH; ++c) {
        p0[c]  = Pb[(size_t)c * L];
        pLv[c] = Pb[(size_t)c * L + (L - 1)];
    }
    // loop-invariant per-lane channel constants: direct scalar loads, no select tree
    const float p0k = Pb[(size_t)k7 * L];
    const float pLk = Pb[(size_t)k7 * L + (L - 1)];

    v8f accS2 = {};
    v8f acc3A[4] = {{}, {}, {}, {}};   // S3 part A: rows i (8), cols jk (64) in 4 N-tiles
    v8f acc3B[4] = {{}, {}, {}, {}};   // S3 part B: rows ij (64) in 4 M-tiles, cols k (8)

    const int perW    = (L + NWAVES - 1) / NWAVES;
    const int wStart  = wave * perW;
    const int nChunks = (perW + TC - 1) / TC;

    for (int ch = 0; ch < nChunks; ++ch) {
        const int gbase = wStart + ch * TC;
        // stage chunk to LDS, time-major rows of 8 floats.
        // chunkBuf[wave] is private to this wave; LDS ops are in-order per wave, so no
        // workgroup barrier is needed around the chunk loop.
        for (int c = 0; c < NCH; ++c) {
            for (int tt = lane; tt < TC + 2; tt += 32) {
                int t = gbase + tt;
                if (t > L - 1) t = L - 1;        // clamped tail -> dx = 0 -> no contribution
                chunkBuf[wave][tt][c] = Pb[(size_t)c * L + t];
            }
        }
        // prefetch next chunk toward the WGP while we compute (global_prefetch_b8)
        if (ch + 1 < nChunks && lane < NCH) {
            int tn = gbase + TC;
            if (tn > L - 1) tn = L - 1;
            __builtin_prefetch(&Pb[(size_t)lane * L + tn], 0, 1);
        }

        for (int g = 0; g < TC; g += 4) {
            const int tta = g + 2 * hi;          // this lane covers increments tta, tta+1
            const float* rowPtr = &chunkBuf[wave][tta][0];

            // channel-k7 scalars: volatile so they stay genuine ds_load_b32 broadcasts
            // (otherwise the compiler forwards from the b128 row loads below and
            //  lowers the dynamic extract as cmp/cndmask select chains)
            volatile const float* vrow = rowPtr;
            float pk0 = vrow[k7];
            float pk1 = vrow[NCH + k7];
            float pk2 = vrow[2 * NCH + k7];
            float dkA  = pk1 - pk0;
            float dkB  = pk2 - pk1;
            float aS2x = (pk0 - p0k) + 0.5f * dkA;   // pre + dx/2 at t
            float aS2y = (pk1 - p0k) + 0.5f * dkB;   // pre + dx/2 at t+1
            float sfx  = pLk - pk1;                  // suf[t]   = pL - p[t+1]
            float sfy  = pLk - pk2;                  // suf[t+1] = pL - p[t+2]

            // full rows for the constant-index tile operands
            const v4f* r0 = (const v4f*)rowPtr;
            v4f q00 = r0[0], q01 = r0[1];
            v4f q10 = r0[2], q11 = r0[3];
            v4f q20 = r0[4], q21 = r0[5];
            float P0[NCH] = {q00.x,q00.y,q00.z,q00.w,q01.x,q01.y,q01.z,q01.w};
            float P1[NCH] = {q10.x,q10.y,q10.z,q10.w,q11.x,q11.y,q11.z,q11.w};
            float P2[NCH] = {q20.x,q20.y,q20.z,q20.w,q21.x,q21.y,q21.z,q21.w};

            float dxA[NCH], dxB[NCH], a2A[NCH];
#pragma unroll
            for (int c = 0; c < NCH; ++c) {
                dxA[c] = P1[c] - P0[c];
                dxB[c] = P2[c] - P1[c];
                a2A[c] = (P0[c] - p0[c]) + 0.5f * dxA[c];   // pre + dx/2
            }

            v2f opA_S2, opA_3A, opB_S2, opB_3B;
            opA_S2.x = mlo ? aS2x : 0.f;
            opA_S2.y = mlo ? aS2y : 0.f;
            opA_3A.x = mlo ? (aS2x - (1.f/6.f) * dkA) : 0.f;   // pre + dx/3 = a2 - dx/6
            opA_3A.y = mlo ? (aS2y - (1.f/6.f) * dkB) : 0.f;
            opB_S2.x = mlo ? dkA : 0.f;
            opB_S2.y = mlo ? dkB : 0.f;
            opB_3B.x = mlo ? sfx : 0.f;
            opB_3B.y = mlo ? sfy : 0.f;

            v2f opB3A[4], opA3B[4];
#pragma unroll
            for (int nt = 0; nt < 4; ++nt) {
                // B tile nt for part A: col jk = nt*16 + m -> j = nt*2 + (m>>3), k = m&7
                float jx = mb ? dxA[2*nt+1] : dxA[2*nt];
                float jy = mb ? dxB[2*nt+1] : dxB[2*nt];
                opB3A[nt].x = 0.5f * jx * dkA;
                opB3A[nt].y = 0.5f * jy * dkB;
                // A tile nt for part B: row ij = nt*16 + m -> i = nt*2 + (m>>3), j = m&7
                // a2 at t+1 = a2 at t + (dx[t]+dx[t+1])/2, reusing jx/jy
                float ix = mb ? a2A[2*nt+1] : a2A[2*nt];
                float iy = ix + 0.5f * (jx + jy);
                opA3B[nt].x = ix * dkA;
                opA3B[nt].y = iy * dkB;
            }

            accS2 = __builtin_amdgcn_wmma_f32_16x16x4_f32(false, opA_S2, false, opB_S2,
                                                          (short)0, accS2, false, false);
#pragma unroll
            for (int nt = 0; nt < 4; ++nt)
                acc3A[nt] = __builtin_amdgcn_wmma_f32_16x16x4_f32(false, opA_3A, false, opB3A[nt],
                                                                 (short)0, acc3A[nt], false, false);
#pragma unroll
            for (int mt = 0; mt < 4; ++mt)
                acc3B[mt] = __builtin_amdgcn_wmma_f32_16x16x4_f32(false, opA3B[mt], false, opB_3B,
                                                                 (short)0, acc3B[mt], false, false);
        }
    }

    // ---- combine accumulators deterministically through LDS ----
    // Phase 1: S2 tile + S3 part A (each a bijection onto its region)
    if (hi == 0) {
        if (mlo) {
#pragma unroll
            for (int r = 0; r < 8; ++r)
                partial[wave][r * 8 + m] = accS2[r];              // S2[r][m]
        }
#pragma unroll
        for (int nt = 0; nt < 4; ++nt)
#pragma unroll
            for (int r = 0; r < 8; ++r)
                partial[wave][64 + r * 64 + nt * 16 + m] = acc3A[nt][r];  // S3[i=r][jk]
    }
    __syncthreads();
    // Phase 2: add S3 part B (bijection onto [64, 576), RMW safe)
    if (mlo) {
#pragma unroll
        for (int mt = 0; mt < 4; ++mt)
#pragma unroll
            for (int r = 0; r < 8; ++r) {
                int ij = mt * 16 + r + 8 * hi;
                partial[wave][64 + ij * 8 + m] += acc3B[mt][r];   // S3[ij][k=m]
            }
    }
    __syncthreads();

    float* ob = out + (size_t)b * SIGDIM;
    if (tid < 8)
        ob[tid] = SEL8(pLv, tid) - SEL8(p0, tid);                 // level 1
    for (int idx = tid; idx < PART; idx += NWAVES * 32)
        ob[8 + idx] = partial[0][idx] + partial[1][idx] +
                      partial[2][idx] + partial[3][idx];
}

extern "C" void kernel_launch(void* const* d_in, const int* in_sizes, int n_in,
                              void* d_out, int out_size, void* d_ws, size_t ws_size,
                              hipStream_t stream)
{
    (void)n_in; (void)d_ws; (void)ws_size;
    const float* path = (const float*)d_in[0];   // (B, 8, L) fp32
    float* out = (float*)d_out;                  // (B, 584) fp32
    int B = out_size / SIGDIM;
    if (B <= 0) return;
    int L = in_sizes[0] / (B * NCH);
    sig3_wmma_kernel<<<B, NWAVES * 32, 0, stream>>>(path, out, L);
}